// GRNNLM_85203561218243
// MI455X (gfx1250) — compile-verified
//
#include <hip/hip_runtime.h>
#include <hip/hip_bf16.h>
#include <math.h>

// Problem constants (from reference): V=32000, E=128, H=32, L=64, B=32
#define VSZ 32000
#define ESZ 128
#define HSZ 32
#define LSZ 64
#define BSZ 32

typedef __attribute__((ext_vector_type(2))) float v2f;
typedef __attribute__((ext_vector_type(8))) float v8f;

// ---------------------------------------------------------------------------
// Kernel 1: bidirectional RNN scan. blockIdx.x = direction (0 fwd, 1 bwd).
// 1024 threads = (b, j) pairs, b=tid/32, j=tid%32 (j == lane -> coalesced W).
// Stores the PRE-update hidden state at each position l:
//   fwd:  hcat[l][b][0:32]  = h before consuming token l
//   bwd:  hcat[l][b][32:64] = h before consuming token l (scanning l=L-1..0)
// which exactly matches hf_used / hb_used in the reference.
// ---------------------------------------------------------------------------
__global__ void __launch_bounds__(1024) rnn_scan_kernel(
    const int* __restrict__ inp, const float* __restrict__ we,
    const float* __restrict__ i2h1, const float* __restrict__ i2h2,
    const float* __restrict__ bias, const float* __restrict__ h0,
    float* __restrict__ hcat)
{
    const int dir = blockIdx.x;
    const int tid = threadIdx.x;
    const int b = tid >> 5;
    const int j = tid & 31;
    const float* __restrict__ W = dir ? i2h2 : i2h1;   // (E+H, H) row-major

    __shared__ float sh[BSZ * HSZ];      // hidden state h[b][j]
    __shared__ float se[BSZ * ESZ];      // embedding tile e[b][i]

    sh[b * HSZ + j] = h0[j];
    const float bj = bias[j];
    __syncthreads();

    for (int s = 0; s < LSZ; ++s) {
        const int l = dir ? (LSZ - 1 - s) : s;

        // emit pre-update state
        hcat[((size_t)l * BSZ + b) * (2 * HSZ) + dir * HSZ + j] = sh[b * HSZ + j];

        // cooperative embedding gather: 32x128 floats, 4 per thread
        #pragma unroll
        for (int k = 0; k < 4; ++k) {
            const int idx = tid + k * 1024;
            const int bb = idx >> 7;
            const int cc = idx & (ESZ - 1);
            se[idx] = we[(size_t)inp[l * BSZ + bb] * ESZ + cc];
        }
        __syncthreads();

        float acc = bj;
        const float* eb = &se[b * ESZ];
        #pragma unroll 8
        for (int i = 0; i < ESZ; ++i)
            acc = fmaf(eb[i], W[i * HSZ + j], acc);
        const float* hb = &sh[b * HSZ];
        #pragma unroll
        for (int i = 0; i < HSZ; ++i)
            acc = fmaf(hb[i], W[(ESZ + i) * HSZ + j], acc);

        const float hn = tanhf(acc);
        __syncthreads();           // all dot-products done before overwrite
        sh[b * HSZ + j] = hn;
        __syncthreads();
    }
}

// ---------------------------------------------------------------------------
// Kernel 2: logits = hcat(2048x64) @ h2o(64x32000), fp32 WMMA 16x16x4.
// One wave per 16x16 C tile; 8 waves/block cover a 16x128 output strip.
// grid = (VSZ/128 = 250, (L*B)/16 = 128), block = 256 (8 wave32s).
//
// Fragment layouts per CDNA5 ISA 7.12.2 (32-bit, wave32):
//   A 16x4 : lane -> m = lane%16 ; vgpr v -> k = 2*(lane/16) + v
//   B 4x16 : lane -> n = lane%16 ; vgpr v -> k = 2*(lane/16) + v
//   C 16x16: vgpr r, lanes 0-15 -> (M=r, N=lane); lanes 16-31 -> (M=r+8)
// ---------------------------------------------------------------------------
__global__ void __launch_bounds__(256) logits_wmma_kernel(
    const float* __restrict__ hcat, const float* __restrict__ h2o,
    float* __restrict__ out)
{
    const int lane = threadIdx.x & 31;
    const int wave = threadIdx.x >> 5;
    const int half = lane >> 4;          // 0 or 1
    const int l16  = lane & 15;

    const int m0 = blockIdx.y * 16;
    const int n0 = (blockIdx.x * 8 + wave) * 16;

    const float* __restrict__ arow = hcat + (size_t)(m0 + l16) * (2 * HSZ);
    const float* __restrict__ bcol = h2o + n0 + l16;

    v8f acc = {};
    #pragma unroll
    for (int kb = 0; kb < 2 * HSZ; kb += 4) {
        const int k = kb + half * 2;
        v2f a = *(const v2f*)(arow + k);                 // A[m][k], A[m][k+1]
        v2f b;
        b.x = bcol[(size_t)k * VSZ];                      // B[k][n]
        b.y = bcol[(size_t)(k + 1) * VSZ];                // B[k+1][n]
        acc = __builtin_amdgcn_wmma_f32_16x16x4_f32(
            /*neg_a=*/false, a, /*neg_b=*/false, b,
            /*c_mod=*/(short)0, acc, /*reuse_a=*/false, /*reuse_b=*/false);
    }

    float* __restrict__ orow = out + (size_t)(m0 + half * 8) * VSZ + n0 + l16;
    #pragma unroll
    for (int r = 0; r < 8; ++r)
        orow[(size_t)r * VSZ] = acc[r];
}

// ---------------------------------------------------------------------------
// Kernel 3: in-place row-wise log_softmax. One block per row (2048 rows),
// 1024 threads, 32 values cached per thread in registers -> one read + one
// write of the 262 MB logits. wave32 shuffle + LDS tree reductions.
// ---------------------------------------------------------------------------
__global__ void __launch_bounds__(1024) logsoftmax_kernel(float* __restrict__ out)
{
    float* __restrict__ p = out + (size_t)blockIdx.x * VSZ;
    const int t = threadIdx.x;
    const int wid = t >> 5, ln = t & 31;

    float vals[32];
    float mx = -INFINITY;
    #pragma unroll
    for (int k = 0; k < 32; ++k) {
        const int i = t + k * 1024;
        const float v = (i < VSZ) ? p[i] : -INFINITY;
        vals[k] = v;
        mx = fmaxf(mx, v);
    }

    __shared__ float redm[32];
    __shared__ float reds[32];

    #pragma unroll
    for (int off = 16; off > 0; off >>= 1)
        mx = fmaxf(mx, __shfl_xor(mx, off, 32));
    if (ln == 0) redm[wid] = mx;
    __syncthreads();
    if (wid == 0) {
        float m = redm[ln];
        #pragma unroll
        for (int off = 16; off > 0; off >>= 1)
            m = fmaxf(m, __shfl_xor(m, off, 32));
        redm[ln] = m;
    }
    __syncthreads();
    mx = redm[0];

    float sum = 0.0f;
    #pragma unroll
    for (int k = 0; k < 32; ++k) {
        const int i = t + k * 1024;
        if (i < VSZ) sum += expf(vals[k] - mx);
    }
    #pragma unroll
    for (int off = 16; off > 0; off >>= 1)
        sum += __shfl_xor(sum, off, 32);
    if (ln == 0) reds[wid] = sum;
    __syncthreads();
    if (wid == 0) {
        float s = reds[ln];
        #pragma unroll
        for (int off = 16; off > 0; off >>= 1)
            s += __shfl_xor(s, off, 32);
        reds[ln] = s;
    }
    __syncthreads();
    const float lse = mx + logf(reds[0]);

    #pragma unroll
    for (int k = 0; k < 32; ++k) {
        const int i = t + k * 1024;
        if (i < VSZ) p[i] = vals[k] - lse;
    }
}

// ---------------------------------------------------------------------------
extern "C" void kernel_launch(void* const* d_in, const int* in_sizes, int n_in,
                              void* d_out, int out_size, void* d_ws, size_t ws_size,
                              hipStream_t stream) {
    const int*   inp  = (const int*)  d_in[0];   // (L,B)
    const float* we   = (const float*)d_in[1];   // (V,E)
    const float* i2h1 = (const float*)d_in[2];   // (E+H,H)
    const float* i2h2 = (const float*)d_in[3];   // (E+H,H)
    const float* h2o  = (const float*)d_in[4];   // (2H,V)
    const float* bias = (const float*)d_in[5];   // (1,H)
    const float* h0   = (const float*)d_in[6];   // (1,H)
    float* out  = (float*)d_out;                 // (L,B,V)
    float* hcat = (float*)d_ws;                  // (L,B,2H) = 512 KB scratch

    rnn_scan_kernel<<<dim3(2), dim3(1024), 0, stream>>>(
        inp, we, i2h1, i2h2, bias, h0, hcat);

    logits_wmma_kernel<<<dim3(VSZ / 128, (LSZ * BSZ) / 16), dim3(256), 0, stream>>>(
        hcat, h2o, out);

    logsoftmax_kernel<<<dim3(LSZ * BSZ), dim3(1024), 0, stream>>>(out);
}